// GCN_Graph_3753801416996
// MI455X (gfx1250) — compile-verified
//
#include <hip/hip_runtime.h>
#include <math.h>

#define NNODES 50000
#define NEDGES 800000
#define NGRAPH 2000
#define HDIM   256
#define ODIM   128
#define NLAYER 4
#define EPSV   1e-5f
#define MTILES (NNODES / 16)   // 3125, exact

typedef __attribute__((ext_vector_type(2))) float v2f;
typedef __attribute__((ext_vector_type(8))) float v8f;

__device__ __constant__ int ATOM_OFF[9] = {0, 119, 123, 135, 147, 157, 163, 169, 171};

// ---------------------------------------------------------------- utilities
__global__ void zero_kernel(float* __restrict__ p, size_t n) {
    size_t i = (size_t)blockIdx.x * blockDim.x + threadIdx.x;
    if (i < n) p[i] = 0.0f;
}

// ------------------------------------------------------------ atom encoder
// h[i][c] = sum_t emb[x[i][t] + off[t]][c]
__global__ void atom_encoder(const int* __restrict__ x,
                             const float* __restrict__ emb,
                             float* __restrict__ h) {
    const int i = blockIdx.x;
    const int c = threadIdx.x;
    float s = 0.0f;
#pragma unroll
    for (int t = 0; t < 9; ++t) {
        const int idx = x[i * 9 + t] + ATOM_OFF[t];
        s += emb[(size_t)idx * HDIM + c];
    }
    h[(size_t)i * HDIM + c] = s;
}

// ---------------------------------------------------------- degree / norm
__global__ void deg_init(float* __restrict__ deg) {
    int i = blockIdx.x * blockDim.x + threadIdx.x;
    if (i < NNODES) deg[i] = 1.0f;  // self loop
}

__global__ void deg_edges(const int* __restrict__ ei, float* __restrict__ deg) {
    int e = blockIdx.x * blockDim.x + threadIdx.x;
    if (e < NEDGES) atomicAdd(&deg[ei[NEDGES + e]], 1.0f);  // col = targets
}

__global__ void dinv_kernel(const float* __restrict__ deg, float* __restrict__ dinv) {
    int i = blockIdx.x * blockDim.x + threadIdx.x;
    if (i < NNODES) {
        float d = deg[i];
        dinv[i] = (d > 0.0f) ? rsqrtf(d) : 0.0f;
    }
}

// ------------------------------------------------------------- WMMA GEMM
// T[N,256] = A[N,256] @ W[256,256], fp32 in / fp32 accumulate.
// Each block: 8 waves, each wave one 16x64 output tile (same 64 W-columns for
// all waves). The 256x64 W panel is staged once in LDS, pair-interleaved so a
// B fragment (K even/odd row pair) is a single ds_load_b64.
__global__ void __launch_bounds__(256) gemm_wmma(const float* __restrict__ A,
                                                 const float* __restrict__ W,
                                                 float* __restrict__ T) {
    // lw[(k>>1)*128 + n*2 + (k&1)] = W[k][ng*64 + n];  128 kpairs x 128 floats = 64KB
    __shared__ float lw[128 * 128];

    const int tid = threadIdx.x;
    const int ng  = blockIdx.y;  // 0..3, 64 columns each

    // cooperative stage: 16384 elements, 64 per thread, coalesced over n
    for (int idx = tid; idx < 256 * 64; idx += 256) {
        const int k = idx >> 6;   // 0..255
        const int n = idx & 63;
        lw[(k >> 1) * 128 + n * 2 + (k & 1)] = W[(size_t)k * HDIM + ng * 64 + n];
    }
    __syncthreads();

    const int lane  = tid & 31;
    const int wave  = tid >> 5;
    const int mtile = blockIdx.x * 8 + wave;
    if (mtile < MTILES) {
        const int m    = lane & 15;
        const int half = lane >> 4;           // 0 or 1 (upper lane half)
        const int row  = mtile * 16 + m;
        const int ncol = ng * 64 + m;         // base column for D fragments

        v8f acc0 = {}, acc1 = {}, acc2 = {}, acc3 = {};

        const float* Arow = A + (size_t)row * HDIM;
        for (int k0 = 0; k0 < HDIM; k0 += 4) {
            // A fragment: lane holds A[row][k0 + 2*half .. +1] (one b64 load)
            const v2f a = *(const v2f*)(Arow + k0 + 2 * half);
            // B fragments from LDS: kpair = k0/2 + half, col = j*16 + m
            const float* lrow = lw + ((k0 >> 1) + half) * 128 + m * 2;
            const v2f b0 = *(const v2f*)(lrow + 0);
            const v2f b1 = *(const v2f*)(lrow + 32);
            const v2f b2 = *(const v2f*)(lrow + 64);
            const v2f b3 = *(const v2f*)(lrow + 96);

            acc0 = __builtin_amdgcn_wmma_f32_16x16x4_f32(false, a, false, b0, (short)0, acc0, false, false);
            acc1 = __builtin_amdgcn_wmma_f32_16x16x4_f32(false, a, false, b1, (short)0, acc1, false, false);
            acc2 = __builtin_amdgcn_wmma_f32_16x16x4_f32(false, a, false, b2, (short)0, acc2, false, false);
            acc3 = __builtin_amdgcn_wmma_f32_16x16x4_f32(false, a, false, b3, (short)0, acc3, false, false);
        }

        // D layout: VGPR v, lane l -> row (v + 8*half), col (l&15)
#pragma unroll
        for (int v = 0; v < 8; ++v) {
            const int r = mtile * 16 + v + 8 * half;
            float* dst = T + (size_t)r * HDIM + ncol;
            dst[0]  = acc0[v];
            dst[16] = acc1[v];
            dst[32] = acc2[v];
            dst[48] = acc3[v];
        }
    }
}

// --------------------------------------------------- propagate (scatter-add)
__global__ void bias_init(const float* __restrict__ b, float* __restrict__ h) {
    const int i = blockIdx.x;
    const int c = threadIdx.x;
    h[(size_t)i * HDIM + c] = b[c];
}

__global__ void scatter_edges(const int* __restrict__ ei,
                              const float* __restrict__ dinv,
                              const float* __restrict__ T,
                              float* __restrict__ h) {
    const int e   = blockIdx.x;
    const int src = ei[e];            // row (source)
    const int dst = ei[NEDGES + e];   // col (target)
    const float nrm = dinv[src] * dinv[dst];
    const int c = threadIdx.x;
    atomicAdd(&h[(size_t)dst * HDIM + c], T[(size_t)src * HDIM + c] * nrm);
}

__global__ void scatter_self(const float* __restrict__ dinv,
                             const float* __restrict__ T,
                             float* __restrict__ h) {
    const int i = blockIdx.x;
    const float d = dinv[i];
    const float nrm = d * d;
    const int c = threadIdx.x;
    atomicAdd(&h[(size_t)i * HDIM + c], T[(size_t)i * HDIM + c] * nrm);
}

// ----------------------------------------------------------- batch norm
__global__ void bn_stats(const float* __restrict__ h,
                         float* __restrict__ sum, float* __restrict__ sq) {
    const int c = threadIdx.x;
    float s = 0.0f, q = 0.0f;
    for (int i = blockIdx.x; i < NNODES; i += gridDim.x) {
        const float v = h[(size_t)i * HDIM + c];
        s += v;
        q += v * v;
    }
    atomicAdd(&sum[c], s);
    atomicAdd(&sq[c], q);
}

__global__ void bn_finalize(const float* __restrict__ sum, const float* __restrict__ sq,
                            const float* __restrict__ gamma, const float* __restrict__ beta,
                            float* __restrict__ scale, float* __restrict__ shift) {
    const int c = threadIdx.x;
    const float mean = sum[c] * (1.0f / NNODES);
    const float var  = sq[c] * (1.0f / NNODES) - mean * mean;
    const float inv  = rsqrtf(var + EPSV);
    const float sc   = gamma[c] * inv;
    scale[c] = sc;
    shift[c] = beta[c] - mean * sc;
}

__global__ void bn_apply_relu(const float* __restrict__ scale,
                              const float* __restrict__ shift,
                              float* __restrict__ h) {
    const int i = blockIdx.x;
    const int c = threadIdx.x;
    const size_t o = (size_t)i * HDIM + c;
    const float v = h[o] * scale[c] + shift[c];
    h[o] = v > 0.0f ? v : 0.0f;
}

// ------------------------------------------------------------- pooling/head
__global__ void pool_accum(const int* __restrict__ batch,
                           const float* __restrict__ h,
                           float* __restrict__ pooled,
                           float* __restrict__ cnt) {
    const int i = blockIdx.x;
    const int g = batch[i];
    const int c = threadIdx.x;
    atomicAdd(&pooled[(size_t)g * HDIM + c], h[(size_t)i * HDIM + c]);
    if (c == 0) atomicAdd(&cnt[g], 1.0f);
}

__global__ void head_kernel(const float* __restrict__ pooled,
                            const float* __restrict__ cnt,
                            const float* __restrict__ lw,
                            const float* __restrict__ lb,
                            float* __restrict__ out) {
    const int g = blockIdx.x;
    const int o = threadIdx.x;  // 0..127
    const float inv = 1.0f / fmaxf(cnt[g], 1.0f);
    float acc = lb[o];
    const float* pg = pooled + (size_t)g * HDIM;
    for (int c = 0; c < HDIM; ++c)
        acc += pg[c] * inv * lw[(size_t)c * ODIM + o];
    out[(size_t)g * ODIM + o] = 1.0f / (1.0f + expf(-acc));
}

// ---------------------------------------------------------------- launcher
extern "C" void kernel_launch(void* const* d_in, const int* in_sizes, int n_in,
                              void* d_out, int out_size, void* d_ws, size_t ws_size,
                              hipStream_t stream) {
    (void)in_sizes; (void)n_in; (void)out_size; (void)ws_size;

    const int*   x_atoms = (const int*)d_in[0];
    const int*   ei      = (const int*)d_in[1];
    const int*   batch   = (const int*)d_in[2];
    const float* emb     = (const float*)d_in[3];
    const float* W_convs = (const float*)d_in[4];
    const float* b_convs = (const float*)d_in[5];
    const float* bn_g    = (const float*)d_in[6];
    const float* bn_b    = (const float*)d_in[7];
    const float* lin_W   = (const float*)d_in[8];
    const float* lin_b   = (const float*)d_in[9];
    float*       out     = (float*)d_out;

    // workspace carve-up (floats)
    float* h      = (float*)d_ws;                        // N*H
    float* t      = h + (size_t)NNODES * HDIM;           // N*H
    float* deg    = t + (size_t)NNODES * HDIM;           // N
    float* dinv   = deg + NNODES;                        // N
    float* bsum   = dinv + NNODES;                       // H
    float* bsq    = bsum + HDIM;                         // H
    float* bscale = bsq + HDIM;                          // H
    float* bshift = bscale + HDIM;                       // H
    float* pooled = bshift + HDIM;                       // G*H
    float* cnt    = pooled + (size_t)NGRAPH * HDIM;      // G

    // encoder
    atom_encoder<<<NNODES, HDIM, 0, stream>>>(x_atoms, emb, h);

    // GCN symmetric norm
    deg_init<<<(NNODES + 255) / 256, 256, 0, stream>>>(deg);
    deg_edges<<<(NEDGES + 255) / 256, 256, 0, stream>>>(ei, deg);
    dinv_kernel<<<(NNODES + 255) / 256, 256, 0, stream>>>(deg, dinv);

    const dim3 gemm_grid((MTILES + 7) / 8, 4);
    for (int l = 0; l < NLAYER; ++l) {
        // t = h @ W[l]   (consumes h completely)
        gemm_wmma<<<gemm_grid, 256, 0, stream>>>(h, W_convs + (size_t)l * HDIM * HDIM, t);
        // h <- b[l]  (destination pre-seeded with bias, then scatter-add)
        bias_init<<<NNODES, HDIM, 0, stream>>>(b_convs + (size_t)l * HDIM, h);
        scatter_edges<<<NEDGES, HDIM, 0, stream>>>(ei, dinv, t, h);
        scatter_self<<<NNODES, HDIM, 0, stream>>>(dinv, t, h);
        if (l < NLAYER - 1) {
            zero_kernel<<<2, 256, 0, stream>>>(bsum, 2 * HDIM);  // bsum & bsq contiguous
            bn_stats<<<256, HDIM, 0, stream>>>(h, bsum, bsq);
            bn_finalize<<<1, HDIM, 0, stream>>>(bsum, bsq,
                                                bn_g + (size_t)l * HDIM,
                                                bn_b + (size_t)l * HDIM,
                                                bscale, bshift);
            bn_apply_relu<<<NNODES, HDIM, 0, stream>>>(bscale, bshift, h);
        }
    }

    // global mean pool + sigmoid head
    const size_t pool_elems = (size_t)NGRAPH * HDIM + NGRAPH;  // pooled & cnt contiguous
    zero_kernel<<<(unsigned)((pool_elems + 255) / 256), 256, 0, stream>>>(pooled, pool_elems);
    pool_accum<<<NNODES, HDIM, 0, stream>>>(batch, h, pooled, cnt);
    head_kernel<<<NGRAPH, ODIM, 0, stream>>>(pooled, cnt, lin_W, lin_b, out);
}